// ConvTemporalGraphical1_1623497638556
// MI455X (gfx1250) — compile-verified
//
#include <hip/hip_runtime.h>

#define N_    64
#define CIN_  192
#define T_    256
#define V_    25
#define K_    3
#define COUT_ 64
#define S_    5
#define OC_   (COUT_*K_)     // 192 = K*COUT
#define TV_   (T_*V_)        // 6400 columns per batch item
#define TB_   8              // timesteps per workgroup
#define NT_   (T_/TB_)       // 32 t-blocks
#define COLS_ (TB_*V_)       // 200 columns per workgroup
#define NCT_  13             // ceil(200/16) 16-wide column tiles
#define MT_   (OC_/16)       // 12 16-row M tiles
#define KV_   (K_*V_)        // 75 contraction depth of graph conv
#define KS2_  19             // ceil(75/4) k-steps for graph conv

typedef __attribute__((ext_vector_type(2))) float v2f;
typedef __attribute__((ext_vector_type(8))) float v8f;

// Kernel 1: fused  y = Wc@x + bc  (WMMA f32, y kept in LDS)  ->
//   (a) per-t-block partial sums of y over t (for the mean-over-T path)
//   (b) graph conv out[n,c,t,w] = sum_{k,v} y[n,k*64+c,t,v] * A[k,v,w]  (also WMMA)
__launch_bounds__(256)
__global__ void fused_gemm_graph(const float* __restrict__ x,
                                 const float* __restrict__ A,
                                 const float* __restrict__ Wc,
                                 const float* __restrict__ bc,
                                 float* __restrict__ out,        // (N,COUT,T,V)
                                 float* __restrict__ ybar_part)  // (N,NT,OC,V)
{
    __shared__ float Ylds[OC_ * COLS_];   // 192 x 200 f32 = 153.6 KB
    __shared__ float Alds[K_ * V_ * V_];  // 1875 f32
    __shared__ float bclds[OC_];          // bias (also pads Alds overreads)

    const int tb   = blockIdx.x;
    const int n    = blockIdx.y;
    const int tid  = threadIdx.x;
    const int wave = tid >> 5;
    const int lane = tid & 31;
    const int lane15 = lane & 15;
    const int hi8  = (lane >> 4) << 3;          // C/D row offset: lanes16-31 hold M+8
    const int koff = (lane < 16) ? 0 : 2;       // A/B frag: lanes16-31 hold K+2

    for (int i = tid; i < K_ * V_ * V_; i += 256) Alds[i] = A[i];
    for (int i = tid; i < OC_; i += 256) bclds[i] = bc[i];
    __syncthreads();

    const float* Xn = x + (size_t)n * CIN_ * TV_;   // [CIN][TV], row stride TV
    const int pbase = tb * COLS_;

    // ---- stage A: 192 x 200 GEMM tile into LDS via v_wmma_f32_16x16x4_f32 ----
    for (int ct = wave; ct < NCT_; ct += 8) {
        const int cb = ct * 16 + lane15;        // column within block (0..207)
        int p = pbase + cb;
        if (p > TV_ - 1) p = TV_ - 1;           // clamp tail tile (results masked on store)

        v8f acc[MT_];
        #pragma unroll
        for (int m = 0; m < MT_; ++m) {
            #pragma unroll
            for (int r = 0; r < 8; ++r) acc[m][r] = bclds[m * 16 + r + hi8];
        }

        auto load_bf = [&](int k, v2f& bf) {
            const float* xp = Xn + (size_t)(k + koff) * TV_ + p;
            bf[0] = xp[0];
            bf[1] = xp[TV_];
        };
        auto load_af = [&](int k, v2f* af) {
            #pragma unroll
            for (int m = 0; m < MT_; ++m) {
                const float* wp = Wc + (size_t)(m * 16 + lane15) * CIN_ + k + koff;
                af[m][0] = wp[0];
                af[m][1] = wp[1];
            }
        };
        auto do_wmma = [&](const v2f* af, v2f bf) {
            #pragma unroll
            for (int m = 0; m < MT_; ++m)
                acc[m] = __builtin_amdgcn_wmma_f32_16x16x4_f32(
                    false, af[m], false, bf, (short)0, acc[m], false, false);
        };

        // Software-pipelined, double-buffered K loop.
        v2f afA[MT_], afB[MT_];
        v2f bfA, bfB;
        load_bf(0, bfA);
        load_af(0, afA);
        for (int k = 0; k < CIN_ - 8; k += 8) {
            load_bf(k + 4, bfB);
            load_af(k + 4, afB);
            do_wmma(afA, bfA);
            load_bf(k + 8, bfA);
            load_af(k + 8, afA);
            do_wmma(afB, bfB);
        }
        load_bf(CIN_ - 4, bfB);
        load_af(CIN_ - 4, afB);
        do_wmma(afA, bfA);
        do_wmma(afB, bfB);

        if (cb < COLS_) {
            #pragma unroll
            for (int m = 0; m < MT_; ++m) {
                #pragma unroll
                for (int r = 0; r < 8; ++r)
                    Ylds[(m * 16 + r + hi8) * COLS_ + cb] = acc[m][r];
            }
        }
    }

    __syncthreads();

    // ---- stage B1: partial T-sums (deterministic: per-(n,tb) slot, no atomics) ----
    {
        float* part = ybar_part + ((size_t)n * NT_ + tb) * (OC_ * V_);
        for (int i = tid; i < OC_ * V_; i += 256) {
            const int o = i / V_, v = i - o * V_;
            float s = 0.f;
            #pragma unroll
            for (int t = 0; t < TB_; ++t) s += Ylds[o * COLS_ + t * V_ + v];
            part[i] = s;
        }
    }

    // ---- stage B2: graph conv as WMMA GEMM ----
    // out[(c,t), w] = sum_kv Ylds[(k*64+c)*COLS + t*25 + v] * Alds[k*625 + v*25 + w]
    // M = 512 (c,t) rows -> 32 tiles (4/wave); N = 25 -> 2 w-tiles; K = 75 -> 19 steps.
    {
        // Hoist adjacency B-fragments (shared by every M-tile of this wave).
        v2f bA0[KS2_], bA1[KS2_];
        #pragma unroll
        for (int ks = 0; ks < KS2_; ++ks) {
            #pragma unroll
            for (int e = 0; e < 2; ++e) {
                const int kv = ks * 4 + koff + e;
                const int kk = (kv >= 50) ? 2 : ((kv >= 25) ? 1 : 0);
                const int vv = kv - kk * 25;
                const int base = kk * (V_ * V_) + vv * V_ + lane15;  // may overread into bclds: discarded cols
                bA0[ks][e] = Alds[base];
                bA1[ks][e] = Alds[base + 16];
            }
        }

        #pragma unroll
        for (int mi = 0; mi < 4; ++mi) {
            const int m0 = (wave + mi * 8) * 16;      // M-tile base row
            const int laneRow = m0 + lane15;
            const int cA = laneRow >> 3;              // c of this lane's A-frag row
            const int tA = laneRow & (TB_ - 1);
            const int ybase = cA * COLS_ + tA * V_;

            v8f acc0 = {}, acc1 = {};
            #pragma unroll
            for (int ks = 0; ks < KS2_; ++ks) {
                v2f yf;
                #pragma unroll
                for (int e = 0; e < 2; ++e) {
                    const int kv = ks * 4 + koff + e;
                    const int kk = (kv >= 50) ? 2 : ((kv >= 25) ? 1 : 0);
                    const int vv = kv - kk * 25;
                    const float val = Ylds[kk * (COUT_ * COLS_) + ybase + vv];
                    yf[e] = (kv < KV_) ? val : 0.f;   // zero-pad kv=75
                }
                acc0 = __builtin_amdgcn_wmma_f32_16x16x4_f32(
                    false, yf, false, bA0[ks], (short)0, acc0, false, false);
                acc1 = __builtin_amdgcn_wmma_f32_16x16x4_f32(
                    false, yf, false, bA1[ks], (short)0, acc1, false, false);
            }

            #pragma unroll
            for (int r = 0; r < 8; ++r) {
                const int pr = m0 + r + hi8;
                const int cc = pr >> 3;
                const int tt = pr & (TB_ - 1);
                float* op = out + (((size_t)n * COUT_ + cc) * T_ + tb * TB_ + tt) * V_;
                op[lane15] = acc0[r];                       // w = 0..15
                if (lane15 < V_ - 16) op[16 + lane15] = acc1[r];  // w = 16..24
            }
        }
    }
}

// Kernel 2: ybar -> x1m/x2m (mean-over-T commutes with the 1x1 convs),
// semantic masked means, graphs output, A passthrough.
__launch_bounds__(256)
__global__ void finalize(const float* __restrict__ ybar_part,
                         const float* __restrict__ A,
                         const int*   __restrict__ node_type,
                         const float* __restrict__ W1, const float* __restrict__ b1,
                         const float* __restrict__ W2, const float* __restrict__ b2,
                         float* __restrict__ outA,   // K*V*V
                         float* __restrict__ outG)   // (N,S,COUT,1,1,V)
{
    __shared__ float ybm[OC_ * V_];        // 4800
    __shared__ float x1s[S_ * COUT_ * V_]; // 8000
    __shared__ float x2s[S_ * COUT_ * V_]; // 8000
    const int n = blockIdx.x;
    const int tid = threadIdx.x;

    if (n == 0) {
        for (int i = tid; i < K_ * V_ * V_; i += 256) outA[i] = A[i];
    }

    const float invT = 1.0f / (float)T_;
    for (int i = tid; i < OC_ * V_; i += 256) {
        float s = 0.f;
        const float* pp = ybar_part + (size_t)n * NT_ * OC_ * V_ + i;
        for (int t = 0; t < NT_; ++t) s += pp[(size_t)t * OC_ * V_];
        ybm[i] = s * invT;
    }
    __syncthreads();

    for (int i = tid; i < S_ * COUT_ * V_; i += 256) {
        const int o2 = i / V_, v = i - o2 * V_;
        float s1 = b1[o2], s2 = b2[o2];
        const float* w1r = W1 + (size_t)o2 * OC_;
        const float* w2r = W2 + (size_t)o2 * OC_;
        for (int o = 0; o < OC_; ++o) {
            const float f = ybm[o * V_ + v];
            s1 += w1r[o] * f;
            s2 += w2r[o] * f;
        }
        x1s[i] = s1;
        x2s[i] = s2;
    }
    __syncthreads();

    for (int j = tid; j < S_ * COUT_; j += 256) {
        const int s = j / COUT_;
        float sum = 0.f, cnt = 0.f;
        #pragma unroll
        for (int v = 0; v < V_; ++v) {
            const bool m = (node_type[v] == s);
            sum += m ? x1s[j * V_ + v] : 0.f;
            cnt += m ? 1.f : 0.f;
        }
        const float sem = sum / cnt;
        float* gp = outG + ((size_t)n * S_ * COUT_ + j) * V_;
        #pragma unroll
        for (int v = 0; v < V_; ++v) gp[v] = sem - x2s[j * V_ + v];
    }
}

extern "C" void kernel_launch(void* const* d_in, const int* in_sizes, int n_in,
                              void* d_out, int out_size, void* d_ws, size_t ws_size,
                              hipStream_t stream) {
    const float* x  = (const float*)d_in[0];
    const float* A  = (const float*)d_in[1];
    const int*   nt = (const int*)d_in[2];
    const float* Wc = (const float*)d_in[3];
    const float* bc = (const float*)d_in[4];
    const float* W1 = (const float*)d_in[5];
    const float* b1 = (const float*)d_in[6];
    const float* W2 = (const float*)d_in[7];
    const float* b2 = (const float*)d_in[8];

    float* out  = (float*)d_out;                          // (N,COUT,T,V)
    float* outA = out + (size_t)N_ * COUT_ * T_ * V_;     // A passthrough
    float* outG = outA + (size_t)K_ * V_ * V_;            // graphs
    float* ybar_part = (float*)d_ws;                      // N*NT*OC*V floats (~39.3 MB)

    dim3 grid1(NT_, N_);
    fused_gemm_graph<<<grid1, 256, 0, stream>>>(x, A, Wc, bc, out, ybar_part);
    finalize<<<N_, 256, 0, stream>>>(ybar_part, A, nt, W1, b1, W2, b2, outA, outG);
}